// Multimodal_Attention_35055523070192
// MI455X (gfx1250) — compile-verified
//
#include <hip/hip_runtime.h>
#include <math.h>

// N = 50000 (multiple of 16), D = 512.
typedef __attribute__((ext_vector_type(2))) float v2f;
typedef __attribute__((ext_vector_type(8))) float v8f;

#define DIM        512
#define TILE_M     16
#define NWAVES     4
#define BLOCK_THR  (NWAVES * 32)
#define ETILES_PER_WAVE 8   // 4 waves * 8 tiles * 16 cols = 512 = DIM

__global__ void mm_att_zero_ws(float* s) {
    if (threadIdx.x < 3) s[threadIdx.x] = 0.0f;
}

__global__ __launch_bounds__(BLOCK_THR)
void mm_att_fused(const float* __restrict__ img, const float* __restrict__ ph,
                  const float* __restrict__ Wsh, const float* __restrict__ bsh,
                  const float* __restrict__ Wim, const float* __restrict__ bim,
                  const float* __restrict__ Wph, const float* __restrict__ bph,
                  float* __restrict__ joint, float* __restrict__ sums)
{
    __shared__ __align__(16) float lds_img[TILE_M * DIM];   // 32 KB
    __shared__ __align__(16) float lds_ph [TILE_M * DIM];   // 32 KB

    const int tid  = threadIdx.x;
    const int wave = tid >> 5;
    const int lane = tid & 31;
    const int n0   = blockIdx.x * TILE_M;

    // Cooperative stage: 16x512 f32 tiles of img and ph -> LDS (float4 traffic).
    {
        const float4* gi = (const float4*)(img + (size_t)n0 * DIM);
        const float4* gp = (const float4*)(ph  + (size_t)n0 * DIM);
        float4* si = (float4*)lds_img;
        float4* sp = (float4*)lds_ph;
        for (int i = tid; i < TILE_M * DIM / 4; i += BLOCK_THR) {
            si[i] = gi[i];
            sp[i] = gp[i];
        }
    }
    __syncthreads();

    // ISA 16x16x4 f32 fragment indexing (wave32):
    //  A: lane holds row M=lane%16, VGPR i -> K = 2*(lane/16) + i
    //  B: lane holds col N=lane%16, VGPR j -> K = 2*(lane/16) + j   (B[k][n] = W[e0+n][k])
    //  C/D: lane, VGPR i -> row M = i + 8*(lane/16), col N = lane%16
    const int nsub  = lane & 15;          // M for A rows / N for B cols & C cols
    const int khalf = (lane >> 4) * 2;    // K sub-offset {0,2}
    const int rbase = (lane >> 4) * 8;    // C/D row base {0,8}

    float psum_img = 0.0f, psum_ph = 0.0f, psum_sh = 0.0f;

    const float* aim_row = lds_img + nsub * DIM;
    const float* aph_row = lds_ph  + nsub * DIM;

    for (int et = 0; et < ETILES_PER_WAVE; ++et) {
        const int e0 = (wave * ETILES_PER_WAVE + et) * 16;

        const float* wim_row = Wim + (size_t)(e0 + nsub) * DIM;
        const float* wph_row = Wph + (size_t)(e0 + nsub) * DIM;
        const float* wsh_row = Wsh + (size_t)(e0 + nsub) * DIM;

        v8f acc_im = {}; v8f acc_ph = {}; v8f acc_sh = {};

        for (int k = 0; k < DIM; k += 4) {
            const int kk = k + khalf;
            v2f a_im = { aim_row[kk], aim_row[kk + 1] };            // ds_load_b64
            v2f a_ph = { aph_row[kk], aph_row[kk + 1] };
            v2f a_sh = 0.5f * (a_im + a_ph);
            v2f b_im = { wim_row[kk], wim_row[kk + 1] };            // global_load_b64 (L2-resident weights)
            v2f b_ph = { wph_row[kk], wph_row[kk + 1] };
            v2f b_sh = { wsh_row[kk], wsh_row[kk + 1] };

            acc_im = __builtin_amdgcn_wmma_f32_16x16x4_f32(
                false, a_im, false, b_im, (short)0, acc_im, false, false);
            acc_ph = __builtin_amdgcn_wmma_f32_16x16x4_f32(
                false, a_ph, false, b_ph, (short)0, acc_ph, false, false);
            acc_sh = __builtin_amdgcn_wmma_f32_16x16x4_f32(
                false, a_sh, false, b_sh, (short)0, acc_sh, false, false);
        }

        // Bias along columns: col index e0 + nsub, identical for all 8 C VGPRs.
        const float bi = bim[e0 + nsub];
        const float bp = bph[e0 + nsub];
        const float bs = bsh[e0 + nsub];

        #pragma unroll
        for (int i = 0; i < 8; ++i) {
            const float ai = tanhf(acc_im[i] + bi);
            const float ap = tanhf(acc_ph[i] + bp);
            const float as = tanhf(acc_sh[i] + bs);
            psum_img += ai * ai;
            psum_ph  += ap * ap;
            psum_sh  += as * as;

            const int m  = rbase + i;                 // row within tile
            const int c  = e0 + nsub;                 // column in [0,512)
            const float xi = lds_img[m * DIM + c];
            const float xp = lds_ph [m * DIM + c];
            const float xs = 0.5f * (xi + xp);
            joint[(size_t)(n0 + m) * DIM + c] = as * xs + ai * xi + ap * xp;
        }
    }

    // Wave-level tree reduction (wave32), then one atomic per wave per sum.
    #pragma unroll
    for (int off = 16; off > 0; off >>= 1) {
        psum_img += __shfl_down(psum_img, off, 32);
        psum_ph  += __shfl_down(psum_ph,  off, 32);
        psum_sh  += __shfl_down(psum_sh,  off, 32);
    }
    if (lane == 0) {
        atomicAdd(&sums[0], psum_img);
        atomicAdd(&sums[1], psum_ph);
        atomicAdd(&sums[2], psum_sh);
    }
}

__global__ void mm_att_finalize(const float* __restrict__ sums, float* __restrict__ w_out) {
    if (threadIdx.x == 0) {
        const float s_img = sums[0];
        const float s_ph  = sums[1];
        const float s_sh  = sums[2];
        const float iscore = s_img / s_sh;
        const float pscore = s_ph  / s_sh;
        const float mx = fmaxf(iscore, pscore);
        const float ei = expf(iscore - mx);
        const float ep = expf(pscore - mx);
        const float inv = 1.0f / (ei + ep);
        w_out[0] = ei * inv;
        w_out[1] = ep * inv;
    }
}

extern "C" void kernel_launch(void* const* d_in, const int* in_sizes, int n_in,
                              void* d_out, int out_size, void* d_ws, size_t ws_size,
                              hipStream_t stream) {
    const float* img = (const float*)d_in[0];   // img_embed (N,512)
    const float* ph  = (const float*)d_in[1];   // ph_embed  (N,512)
    const float* Wsh = (const float*)d_in[2];   // W_shared  (512,512)
    const float* bsh = (const float*)d_in[3];   // b_shared  (512,)
    const float* Wim = (const float*)d_in[4];   // W_img
    const float* bim = (const float*)d_in[5];   // b_img
    const float* Wph = (const float*)d_in[6];   // W_ph
    const float* bph = (const float*)d_in[7];   // b_ph

    const int N = in_sizes[0] / DIM;            // 50000
    float* joint = (float*)d_out;               // N*512 floats
    float* w_out = (float*)d_out + (size_t)N * DIM;  // 2 floats
    float* sums  = (float*)d_ws;                // 3 partial-sum accumulators

    mm_att_zero_ws<<<1, 32, 0, stream>>>(sums);
    mm_att_fused<<<N / TILE_M, BLOCK_THR, 0, stream>>>(
        img, ph, Wsh, bsh, Wim, bim, Wph, bph, joint, sums);
    mm_att_finalize<<<1, 32, 0, stream>>>(sums, w_out);
}